// CMPO3_GTN_61641370632907
// MI455X (gfx1250) — compile-verified
//
#include <hip/hip_runtime.h>

typedef __attribute__((ext_vector_type(2))) float v2f;
typedef __attribute__((ext_vector_type(8))) float v8f;
typedef __attribute__((ext_vector_type(4))) unsigned int u32x4;
typedef __attribute__((ext_vector_type(8))) int i32x8;
typedef __attribute__((ext_vector_type(4))) int i32x4;

#define L_   64
#define CH_  16
#define PIX_ 256
#define PAT_ 64
#define RC_  32
#define BD_  64
#define OUT_ 10
#define B_   256

// ---------------------------------------------------------------------------
// TDM 2D tile load: 16 rows of `tile_d0` f32 elements, row stride
// `row_stride` elements, into LDS at lds_byte_addr with per-row padding
// (pad codes per D# group1 spec: interval 2^(ic+1) dwords, amount ac+1 dwords).
// D# bit packing per CDNA5 ISA 8.3/8.4 (count=1, type=2, data_size=4B).
// This toolchain exposes the 6-arg builtin form (groups 0..3 + extra + cpol).
// ---------------------------------------------------------------------------
__device__ __forceinline__ void tdm_load_tile16(
    unsigned lds_byte_addr, const float* gsrc,
    unsigned tile_d0, unsigned long row_stride,
    unsigned pad_ic, unsigned pad_ac)
{
  unsigned long ga = (unsigned long)gsrc;
  u32x4 g0;
  g0[0] = 0x1u;                                   // count=1 (valid), user mode
  g0[1] = lds_byte_addr;                          // lds_addr
  g0[2] = (unsigned)(ga & 0xffffffffu);           // global_addr[31:0]
  g0[3] = (unsigned)((ga >> 32) & 0x01ffffffu)    // global_addr[56:32]
        | (2u << 30);                             // type=2 ("image")
  i32x8 g1;
  unsigned w0 = (2u << 16)                        // data_size = 4 bytes
              | (1u << 20)                        // pad_enable
              | (pad_ic << 22)                    // pad_interval code
              | (pad_ac << 25);                   // pad_amount code
  unsigned td0 = tile_d0;                         // tensor_dim0 == tile width
  unsigned td1 = 16u;                             // tensor_dim1 == 16 rows
  g1[0] = (int)w0;                                // mask=0 | flags
  g1[1] = (int)((td0 & 0xffffu) << 16);           // [47:32]=0, [63:48]=dim0.lo
  g1[2] = (int)((td0 >> 16) | ((td1 & 0xffffu) << 16)); // dim0.hi | dim1.lo
  g1[3] = (int)((td1 >> 16) | (tile_d0 << 16));   // dim1.hi | tile_dim0
  g1[4] = (int)16;                                // tile_dim1=16, tile_dim2=0
  g1[5] = (int)(unsigned)(row_stride & 0xffffffffu);      // dim0_stride lo
  g1[6] = (int)(unsigned)((row_stride >> 32) & 0xffffu);  // dim0_stride hi
  g1[7] = 0;                                      // dim1_stride = 0 (2D)
  i32x4 zz4 = {0, 0, 0, 0};                       // groups 2/3 unused (2D)
  i32x8 zz8 = {0, 0, 0, 0, 0, 0, 0, 0};           // extra group (unused)
  __builtin_amdgcn_tensor_load_to_lds(g0, g1, zz4, zz4, zz8, 0);
}

// ---------------------------------------------------------------------------
// 1) x (B,PAT,PIX,CH) -> vch[l][b][c] = mean_pix, vpx[l][b][p] = mean_ch
// ---------------------------------------------------------------------------
__global__ void k_reduce_x(const float* __restrict__ x,
                           float* __restrict__ vch,   // [L][B][CH]
                           float* __restrict__ vpx)   // [L][B][PIX]
{
  const int s = blockIdx.x;            // 0 .. B*L-1
  const int b = s / L_;
  const int l = s % L_;
  const int t = threadIdx.x;           // pixel 0..255
  const float* xp = x + (((size_t)b * PAT_ + l) * PIX_ + t) * CH_;

  __shared__ float accC[CH_];
  if (t < CH_) accC[t] = 0.0f;
  __syncthreads();

  float4 a0 = *reinterpret_cast<const float4*>(xp + 0);
  float4 a1 = *reinterpret_cast<const float4*>(xp + 4);
  float4 a2 = *reinterpret_cast<const float4*>(xp + 8);
  float4 a3 = *reinterpret_cast<const float4*>(xp + 12);

  float sumc = (a0.x + a0.y + a0.z + a0.w) + (a1.x + a1.y + a1.z + a1.w) +
               (a2.x + a2.y + a2.z + a2.w) + (a3.x + a3.y + a3.z + a3.w);
  vpx[((size_t)l * B_ + b) * PIX_ + t] = sumc * (1.0f / CH_);

  atomicAdd(&accC[0],  a0.x); atomicAdd(&accC[1],  a0.y);
  atomicAdd(&accC[2],  a0.z); atomicAdd(&accC[3],  a0.w);
  atomicAdd(&accC[4],  a1.x); atomicAdd(&accC[5],  a1.y);
  atomicAdd(&accC[6],  a1.z); atomicAdd(&accC[7],  a1.w);
  atomicAdd(&accC[8],  a2.x); atomicAdd(&accC[9],  a2.y);
  atomicAdd(&accC[10], a2.z); atomicAdd(&accC[11], a2.w);
  atomicAdd(&accC[12], a3.x); atomicAdd(&accC[13], a3.y);
  atomicAdd(&accC[14], a3.z); atomicAdd(&accC[15], a3.w);
  __syncthreads();
  if (t < CH_) vch[((size_t)l * B_ + b) * CH_ + t] = accC[t] * (1.0f / PIX_);
}

// ---------------------------------------------------------------------------
// 2) first site: v0[b][r] = sum_p first[r][p] * u[b][p]
// ---------------------------------------------------------------------------
template <int N, int P>
__global__ void k_first(const float* __restrict__ first,  // [N][P]
                        const float* __restrict__ u,      // [B][P]
                        float* __restrict__ v0)           // [B][N]
{
  const int tid = blockIdx.x * blockDim.x + threadIdx.x;
  if (tid >= B_ * N) return;
  const int b = tid / N, r = tid % N;
  const float* up = u + (size_t)b * P;
  const float* fp = first + (size_t)r * P;
  float s = 0.f;
#pragma unroll 4
  for (int p = 0; p < P; ++p) s += fp[p] * up[p];
  v0[b * N + r] = s;
}

__global__ void k_zero(float* __restrict__ p, int n) {
  const int i = blockIdx.x * blockDim.x + threadIdx.x;
  if (i < n) p[i] = 0.0f;
}

// ---------------------------------------------------------------------------
// 3) chain step: v'[b,r] = sum_{l,p} (v[b,l]*u[b,p]) * W[l,r,p]
//    One block per (l, 16-column slab): W slab [l, n0:n0+16, 0:P] is TDM-DMA'd
//    into LDS ONCE (row-padded so LDS row stride = SLDS == 3 mod 64 dwords ->
//    conflict-free b32 B-frag reads), then reused by 8 waves x 2 M-tiles each
//    covering all 256 batch rows. v[m,l] is loop-invariant (hoisted).
//    V_WMMA_F32_16X16X4_F32, split-(l) partials accumulated with f32 atomics.
// ---------------------------------------------------------------------------
template <int N, int P, int SLDS, int PAD_IC, int PAD_AC>
__global__ void k_chain_tdm(const float* __restrict__ v_in,  // [B][N]
                            const float* __restrict__ W,     // [N][N][P] (l,r,p)
                            const float* __restrict__ u,     // [B][P]
                            float* __restrict__ v_out)       // [B][N], zeroed
{
  constexpr int TN = N / 16;
  const int l  = blockIdx.x / TN;          // bond index handled by this block
  const int tn = blockIdx.x % TN;          // output column tile
  const int n0 = tn * 16;

  __shared__ float wt[16 * SLDS];

  // DMA the 16 x P f32 slab W[l, n0:n0+16, :] into LDS (one TDM op per block).
  if (threadIdx.x == 0) {
    // flat shared pointer low 32 bits == LDS byte offset (ISA 10.2 aperture map)
    unsigned lds_off = (unsigned)(size_t)(void*)&wt[0];
    tdm_load_tile16(lds_off, W + ((size_t)l * N + n0) * P,
                    (unsigned)P, (unsigned long)P, PAD_IC, PAD_AC);
  }
  __builtin_amdgcn_s_wait_tensorcnt((unsigned short)0);
  __syncthreads();

  const int wave = threadIdx.x >> 5;       // 0..7
  const int lane = threadIdx.x & 31;
  const int half = lane >> 4;              // K sub-pair select (ISA A/B layout)
  const int nl   = lane & 15;              // column within tile / row within tile
  const int m0   = (wave + 0) * 16 + nl;   // M-tile 0 row
  const int m1   = (wave + 8) * 16 + nl;   // M-tile 1 row

  const float  vv0 = v_in[(size_t)m0 * N + l];
  const float  vv1 = v_in[(size_t)m1 * N + l];
  const float* __restrict__ urow0 = u + (size_t)m0 * P;
  const float* __restrict__ urow1 = u + (size_t)m1 * P;
  const float* __restrict__ wrow  = wt + nl * SLDS;

  v8f acc0 = {};
  v8f acc1 = {};
#pragma unroll 4
  for (int p0 = 0; p0 < P; p0 += 4) {
    const int pb = p0 + 2 * half;
    // B-frag (4x16): lane holds K={pb,pb+1}, N=nl  (two b32 LDS reads)
    v2f bb; bb.x = wrow[pb]; bb.y = wrow[pb + 1];
    // A-frags (16x4): lane holds M=m, K={pb,pb+1}; z = v[m,l]*u[m,p]
    const float2 u0 = *reinterpret_cast<const float2*>(urow0 + pb);
    const float2 u1 = *reinterpret_cast<const float2*>(urow1 + pb);
    v2f a0; a0.x = vv0 * u0.x; a0.y = vv0 * u0.y;
    v2f a1; a1.x = vv1 * u1.x; a1.y = vv1 * u1.y;
    acc0 = __builtin_amdgcn_wmma_f32_16x16x4_f32(false, a0, false, bb,
                                                 (short)0, acc0, false, false);
    acc1 = __builtin_amdgcn_wmma_f32_16x16x4_f32(false, a1, false, bb,
                                                 (short)0, acc1, false, false);
  }

  const int col = n0 + nl;
#pragma unroll
  for (int i = 0; i < 8; ++i) {
    const int r0 = (wave + 0) * 16 + i + 8 * half;   // C layout: M = i + 8h
    const int r1 = (wave + 8) * 16 + i + 8 * half;
    atomicAdd(&v_out[r0 * N + col], acc0[i]);
    atomicAdd(&v_out[r1 * N + col], acc1[i]);
  }
}

// ---------------------------------------------------------------------------
// 4) psi close: psi_val[b] = sum_p vpx_last[b,p] * sum_l v[b,l]*last[l,p]
// ---------------------------------------------------------------------------
__global__ void k_psi_close(const float* __restrict__ v,         // [B][BD]
                            const float* __restrict__ last,      // [BD][PIX]
                            const float* __restrict__ vpx_last,  // [B][PIX]
                            float* __restrict__ psi_val)         // [B]
{
  const int b = blockIdx.x;
  const int p = threadIdx.x;  // 0..255
  __shared__ float sv[BD_];
  __shared__ float red[256];
  if (p < BD_) sv[p] = v[b * BD_ + p];
  __syncthreads();
  float s = 0.f;
#pragma unroll 8
  for (int l = 0; l < BD_; ++l) s += sv[l] * last[(size_t)l * PIX_ + p];
  red[p] = s * vpx_last[(size_t)b * PIX_ + p];
  __syncthreads();
  for (int st = 128; st > 0; st >>= 1) {
    if (p < st) red[p] += red[p + st];
    __syncthreads();
  }
  if (p == 0) psi_val[b] = red[0];
}

// ---------------------------------------------------------------------------
// 5) phi chain: batch-independent scalar (one-hot phys legs). One block.
// ---------------------------------------------------------------------------
__global__ void k_phi(const float* __restrict__ first,  // [BD][PAT]
                      const float* __restrict__ mid,    // [62][BD][BD][PAT]
                      const float* __restrict__ last,   // [BD][PAT]
                      float* __restrict__ phi_val)
{
  const int t = threadIdx.x;  // 0..63
  __shared__ float ua[BD_], ub[BD_];
  ua[t] = first[t * PAT_ + 0];
  __syncthreads();
  for (int i = 0; i < L_ - 2; ++i) {
    const int p = i + 1;
    float s = 0.f;
    for (int l = 0; l < BD_; ++l)
      s += ua[l] * mid[(((size_t)i * BD_ + l) * BD_ + t) * PAT_ + p];
    ub[t] = s;
    __syncthreads();
    ua[t] = ub[t];
    __syncthreads();
  }
  ua[t] *= last[t * PAT_ + (PAT_ - 1)];
  __syncthreads();
  for (int st = 32; st > 0; st >>= 1) {
    if (t < st) ua[t] += ua[t + st];
    __syncthreads();
  }
  if (t == 0) *phi_val = ua[0];
}

// ---------------------------------------------------------------------------
// 6) chi close + final scale: out[b,o] = chi_out[b,o] * psi_val[b] * phi_val
// ---------------------------------------------------------------------------
__global__ void k_final(const float* __restrict__ v_chi,    // [B][RC]
                        const float* __restrict__ chi_last, // [RC][CH][OUT]
                        const float* __restrict__ vch_last, // [B][CH]
                        const float* __restrict__ psi_val,  // [B]
                        const float* __restrict__ phi_val,  // scalar
                        float* __restrict__ out)            // [B][OUT]
{
  const int b = blockIdx.x;
  const int t = threadIdx.x;  // 0..31 (one wave)
  __shared__ float sv[RC_];
  __shared__ float su[CH_];
  sv[t] = v_chi[b * RC_ + t];
  if (t < CH_) su[t] = vch_last[b * CH_ + t];
  __syncthreads();
  if (t < OUT_) {
    float s = 0.f;
    for (int l = 0; l < RC_; ++l)
      for (int p = 0; p < CH_; ++p)
        s += sv[l] * su[p] * chi_last[((size_t)l * CH_ + p) * OUT_ + t];
    out[b * OUT_ + t] = s * psi_val[b] * phi_val[0];
  }
}

// ---------------------------------------------------------------------------
extern "C" void kernel_launch(void* const* d_in, const int* in_sizes, int n_in,
                              void* d_out, int out_size, void* d_ws, size_t ws_size,
                              hipStream_t stream)
{
  (void)in_sizes; (void)n_in; (void)out_size; (void)ws_size;
  const float* x         = (const float*)d_in[0];
  const float* chi_first = (const float*)d_in[1];
  const float* chi_mid   = (const float*)d_in[2];
  const float* chi_last  = (const float*)d_in[3];
  const float* psi_first = (const float*)d_in[4];
  const float* psi_mid   = (const float*)d_in[5];
  const float* psi_last  = (const float*)d_in[6];
  const float* phi_first = (const float*)d_in[7];
  const float* phi_mid   = (const float*)d_in[8];
  const float* phi_last  = (const float*)d_in[9];
  float* out = (float*)d_out;

  // workspace layout (floats); total ~18 MB
  float* ws = (float*)d_ws;
  size_t off = 0;
  float* vch  = ws + off; off += (size_t)L_ * B_ * CH_;    // 262144
  float* vpx  = ws + off; off += (size_t)L_ * B_ * PIX_;   // 4194304
  float* psiA = ws + off; off += (size_t)B_ * BD_;
  float* psiB = ws + off; off += (size_t)B_ * BD_;
  float* chiA = ws + off; off += (size_t)B_ * RC_;
  float* chiB = ws + off; off += (size_t)B_ * RC_;
  float* psiV = ws + off; off += B_;
  float* phiV = ws + off; off += 64;

  // 1) data reductions
  k_reduce_x<<<B_ * L_, 256, 0, stream>>>(x, vch, vpx);

  // 2) first sites
  k_first<BD_, PIX_><<<(B_ * BD_ + 255) / 256, 256, 0, stream>>>(psi_first, vpx, psiA);
  k_first<RC_, CH_><<<(B_ * RC_ + 255) / 256, 256, 0, stream>>>(chi_first, vch, chiA);

  // 3) sequential chains (62 sites each): per-site TDM+WMMA GEMM.
  //    psi: LDS row stride 259 (=256+3 pad; interval code 7 = 256 dw, amount
  //    code 2 = 3 dw); chi: stride 67 (=16+51; interval code 3 = 16 dw,
  //    amount code 50 = 51 dw). Both == 3 mod 64 -> conflict-free B-frags.
  float* pc = psiA; float* pn = psiB;
  float* cc = chiA; float* cn = chiB;
  for (int i = 0; i < L_ - 2; ++i) {
    k_zero<<<(B_ * BD_ + 255) / 256, 256, 0, stream>>>(pn, B_ * BD_);
    k_chain_tdm<BD_, PIX_, 259, 7, 2><<<BD_ * (BD_ / 16), 256, 0, stream>>>(
        pc, psi_mid + (size_t)i * BD_ * BD_ * PIX_,
        vpx + (size_t)(i + 1) * B_ * PIX_, pn);
    { float* t = pc; pc = pn; pn = t; }

    k_zero<<<(B_ * RC_ + 255) / 256, 256, 0, stream>>>(cn, B_ * RC_);
    k_chain_tdm<RC_, CH_, 67, 3, 50><<<RC_ * (RC_ / 16), 256, 0, stream>>>(
        cc, chi_mid + (size_t)i * RC_ * RC_ * CH_,
        vch + (size_t)(i + 1) * B_ * CH_, cn);
    { float* t = cc; cc = cn; cn = t; }
  }

  // 4-6) closes + final combine
  k_psi_close<<<B_, 256, 0, stream>>>(pc, psi_last,
                                      vpx + (size_t)(L_ - 1) * B_ * PIX_, psiV);
  k_phi<<<1, 64, 0, stream>>>(phi_first, phi_mid, phi_last, phiV);
  k_final<<<B_, 32, 0, stream>>>(cc, chi_last,
                                 vch + (size_t)(L_ - 1) * B_ * CH_, psiV, phiV, out);
}